// GATClassifier_21852793602575
// MI455X (gfx1250) — compile-verified
//
#include <hip/hip_runtime.h>
#include <cmath>

#define N_NODES 100000
#define N_EDGES 1600000
#define NB 64
#define NH 4
#define ND 32
#define NF 128   // NH*ND
#define NCLS 10
#define NEG_SLOPE 0.2f

typedef __attribute__((ext_vector_type(2))) float v2f;
typedef __attribute__((ext_vector_type(8))) float v8f;

// ---------- helpers ----------
__device__ __forceinline__ float atomicMaxF(float* addr, float val) {
  // works for mixed-sign floats when initialized to -FLT_MAX
  if (val >= 0.0f)
    return __int_as_float(atomicMax((int*)addr, __float_as_int(val)));
  else
    return __uint_as_float(atomicMin((unsigned int*)addr, __float_as_uint(val)));
}

__global__ void fill_f32(float* __restrict__ p, float v, int n) {
  int i = blockIdx.x * blockDim.x + threadIdx.x;
  if (i < n) p[i] = v;
}

// ---------- in-degree ----------
__global__ void k_degree(const int* __restrict__ dst, float* __restrict__ deg, int E) {
  int i = blockIdx.x * blockDim.x + threadIdx.x;
  if (i < E) atomicAdd(&deg[dst[i]], 1.0f);
}

// ---------- layer0 node terms: el/er = deg * (W0 . a) ----------
__global__ void k_node0(const float* __restrict__ deg, const float* __restrict__ W0,
                        const float* __restrict__ al0, const float* __restrict__ ar0,
                        float* __restrict__ el, float* __restrict__ er, int N) {
  int i = blockIdx.x * blockDim.x + threadIdx.x;   // i = n*NH + h
  if (i >= N * NH) return;
  int h = i & 3, n = i >> 2;
  float da = 0.f, dr = 0.f;
  #pragma unroll
  for (int d = 0; d < ND; ++d) {
    float w = W0[h * ND + d];
    da += w * al0[h * ND + d];
    dr += w * ar0[h * ND + d];
  }
  float v = deg[n];
  el[i] = v * da;
  er[i] = v * dr;
}

// ---------- generic edge passes ----------
__global__ void k_edge_max(const int* __restrict__ src, const int* __restrict__ dst,
                           const float* __restrict__ el, const float* __restrict__ er,
                           float* __restrict__ m, int E) {
  int e = blockIdx.x * blockDim.x + threadIdx.x;
  if (e >= E) return;
  int s = src[e], d = dst[e];
  float4 a = ((const float4*)el)[s];
  float4 b = ((const float4*)er)[d];
  float x0 = a.x + b.x, x1 = a.y + b.y, x2 = a.z + b.z, x3 = a.w + b.w;
  x0 = x0 > 0.f ? x0 : NEG_SLOPE * x0;
  x1 = x1 > 0.f ? x1 : NEG_SLOPE * x1;
  x2 = x2 > 0.f ? x2 : NEG_SLOPE * x2;
  x3 = x3 > 0.f ? x3 : NEG_SLOPE * x3;
  atomicMaxF(&m[d * 4 + 0], x0);
  atomicMaxF(&m[d * 4 + 1], x1);
  atomicMaxF(&m[d * 4 + 2], x2);
  atomicMaxF(&m[d * 4 + 3], x3);
}

__global__ void k_edge_sum(const int* __restrict__ src, const int* __restrict__ dst,
                           const float* __restrict__ el, const float* __restrict__ er,
                           const float* __restrict__ m, float* __restrict__ ssum, int E) {
  int e = blockIdx.x * blockDim.x + threadIdx.x;
  if (e >= E) return;
  int s = src[e], d = dst[e];
  float4 a = ((const float4*)el)[s];
  float4 b = ((const float4*)er)[d];
  float4 mm = ((const float4*)m)[d];
  float x0 = a.x + b.x, x1 = a.y + b.y, x2 = a.z + b.z, x3 = a.w + b.w;
  x0 = x0 > 0.f ? x0 : NEG_SLOPE * x0;
  x1 = x1 > 0.f ? x1 : NEG_SLOPE * x1;
  x2 = x2 > 0.f ? x2 : NEG_SLOPE * x2;
  x3 = x3 > 0.f ? x3 : NEG_SLOPE * x3;
  atomicAdd(&ssum[d * 4 + 0], __expf(x0 - mm.x));
  atomicAdd(&ssum[d * 4 + 1], __expf(x1 - mm.y));
  atomicAdd(&ssum[d * 4 + 2], __expf(x2 - mm.z));
  atomicAdd(&ssum[d * 4 + 3], __expf(x3 - mm.w));
}

// ---------- layer0 aggregation: z[src] = deg[src]*W0 (rank-1) ----------
__global__ void k_aggr0(const int* __restrict__ src, const int* __restrict__ dst,
                        const float* __restrict__ el, const float* __restrict__ er,
                        const float* __restrict__ m, const float* __restrict__ ssum,
                        const float* __restrict__ deg, const float* __restrict__ W0,
                        float* __restrict__ out, int E) {
  long long t = (long long)blockIdx.x * blockDim.x + threadIdx.x;
  if (t >= (long long)E * NF) return;
  int e = (int)(t >> 7);
  int j = (int)(t & 127);
  int h = j >> 5;
  int s = src[e], d = dst[e];
  float x = el[s * 4 + h] + er[d * 4 + h];
  x = x > 0.f ? x : NEG_SLOPE * x;
  float ex = __expf(x - m[d * 4 + h]);
  float alpha = ex / (ssum[d * 4 + h] + 1e-9f);
  float z = deg[s] * W0[j];
  atomicAdd(&out[(long long)d * NF + j], alpha * z);
}

// ---------- layer1 aggregation: gather z1[src] ----------
__global__ void k_aggr1(const int* __restrict__ src, const int* __restrict__ dst,
                        const float* __restrict__ el, const float* __restrict__ er,
                        const float* __restrict__ m, const float* __restrict__ ssum,
                        const float* __restrict__ z1, float* __restrict__ out, int E) {
  long long t = (long long)blockIdx.x * blockDim.x + threadIdx.x;
  if (t >= (long long)E * NF) return;
  int e = (int)(t >> 7);
  int j = (int)(t & 127);
  int h = j >> 5;
  int s = src[e], d = dst[e];
  float x = el[s * 4 + h] + er[d * 4 + h];
  x = x > 0.f ? x : NEG_SLOPE * x;
  float ex = __expf(x - m[d * 4 + h]);
  float alpha = ex / (ssum[d * 4 + h] + 1e-9f);
  float z = z1[(long long)s * NF + j];
  atomicAdd(&out[(long long)d * NF + j], alpha * z);
}

// ---------- elementwise ELU (in place) ----------
__global__ void k_elu(float* __restrict__ x, int n) {
  int i = blockIdx.x * blockDim.x + threadIdx.x;
  if (i >= n) return;
  float v = x[i];
  x[i] = v > 0.f ? v : (__expf(v) - 1.0f);
}

// ---------- layer1 GEMM: z1[N,128] = h1[N,128] @ Wall[128,128] via WMMA f32 16x16x4 ----------
// Wall[f][h*32+d] = W1[h][f][d]. Grid-stride over 16-row tiles; W staged in LDS once per
// block, A tile staged per tile. Padded LDS strides for conflict-free ds reads.
#define AST 132   // A tile LDS row stride (floats): 132 % 64 == 4 -> b64 reads conflict-free
#define BST 144   // W LDS row stride (floats): 144 % 64 == 16 -> half-waves hit disjoint banks
__global__ __launch_bounds__(256) void k_gemm_wmma(const float* __restrict__ h1,
                                                   const float* __restrict__ W1,
                                                   float* __restrict__ z1, int N) {
  __shared__ float Wl[NF * BST];   // 73728 B
  __shared__ float Al[16 * AST];   // 8448 B
  int tid = threadIdx.x;

  // stage W (transpose [H,128,32] -> Wall[f][h*32+d]) once per block
  for (int i = tid; i < NF * NF; i += 256) {
    int h = i >> 12;          // i = h*4096 + f*32 + d
    int f = (i >> 5) & 127;
    int d = i & 31;
    Wl[f * BST + h * ND + d] = W1[i];
  }

  int wave = tid >> 5;            // col tile 0..7
  int lane = tid & 31;
  int arow = lane & 15;           // A: M row for this lane
  int koff = (lane >> 4) * 2;     // A/B: K pair selected by half-wave
  int colw = wave * 16 + (lane & 15);

  // A staging coords: thread -> (row, 8-float chunk)
  int sr = tid >> 4;              // 0..15
  int sc = (tid & 15) * 8;        // 0..120

  int ntiles = (N + 15) / 16;
  for (int t = blockIdx.x; t < ntiles; t += gridDim.x) {
    int row0 = t * 16;
    __syncthreads();   // Al safe to overwrite (also orders W staging on first pass)
    {
      int n = row0 + sr;
      float4 v0 = {0.f, 0.f, 0.f, 0.f}, v1 = {0.f, 0.f, 0.f, 0.f};
      if (n < N) {
        const float4* p = (const float4*)(h1 + (long long)n * NF + sc);
        v0 = p[0];
        v1 = p[1];
      }
      *(float4*)&Al[sr * AST + sc]     = v0;
      *(float4*)&Al[sr * AST + sc + 4] = v1;
    }
    __syncthreads();

    v8f acc = {};
    #pragma unroll
    for (int k = 0; k < NF; k += 4) {
      v2f a = *(const v2f*)&Al[arow * AST + k + koff];  // one ds_load_b64
      v2f b;
      b.x = Wl[(k + koff) * BST + colw];                 // ds_load_2addr pair
      b.y = Wl[(k + koff + 1) * BST + colw];
      acc = __builtin_amdgcn_wmma_f32_16x16x4_f32(
          /*neg_a=*/false, a, /*neg_b=*/false, b,
          /*c_mod=*/(short)0, acc, /*reuse_a=*/false, /*reuse_b=*/false);
    }

    int rbase = row0 + (lane >> 4) * 8;
    #pragma unroll
    for (int j = 0; j < 8; ++j) {
      int r = rbase + j;
      if (r < N) z1[(long long)r * NF + colw] = acc[j];
    }
  }
}

// ---------- layer1 node terms from z1 ----------
__global__ void k_node1(const float* __restrict__ z1, const float* __restrict__ al1,
                        const float* __restrict__ ar1, float* __restrict__ el,
                        float* __restrict__ er, int N) {
  int i = blockIdx.x * blockDim.x + threadIdx.x;   // n*NH + h
  if (i >= N * NH) return;
  int h = i & 3, n = i >> 2;
  const float* z = z1 + (long long)n * NF + h * ND;
  float a = 0.f, b = 0.f;
  #pragma unroll
  for (int d = 0; d < ND; ++d) {
    float zv = z[d];
    a += zv * al1[h * ND + d];
    b += zv * ar1[h * ND + d];
  }
  el[i] = a;
  er[i] = b;
}

// ---------- mean pooling over graphs ----------
__global__ void k_pool(const float* __restrict__ h2, const int* __restrict__ gid,
                       float* __restrict__ hg, float* __restrict__ cnt, int N) {
  long long t = (long long)blockIdx.x * blockDim.x + threadIdx.x;
  if (t >= (long long)N * NF) return;
  int n = (int)(t >> 7);
  int j = (int)(t & 127);
  int g = gid[n];
  atomicAdd(&hg[g * NF + j], h2[t]);
  if (j == 0) atomicAdd(&cnt[g], 1.0f);
}

// ---------- classifier ----------
__global__ void k_classify(const float* __restrict__ hg, const float* __restrict__ cnt,
                           const float* __restrict__ Wc, const float* __restrict__ bc,
                           float* __restrict__ out) {
  int t = blockIdx.x * blockDim.x + threadIdx.x;
  if (t >= NB * NCLS) return;
  int b = t / NCLS, c = t % NCLS;
  float inv = 1.0f / (cnt[b] + 1e-9f);
  float acc = bc[c];
  #pragma unroll 8
  for (int f = 0; f < NF; ++f) acc += hg[b * NF + f] * inv * Wc[f * NCLS + c];
  out[t] = acc;
}

static inline unsigned cdiv(long long n, int b) { return (unsigned)((n + b - 1) / b); }

extern "C" void kernel_launch(void* const* d_in, const int* in_sizes, int n_in,
                              void* d_out, int out_size, void* d_ws, size_t ws_size,
                              hipStream_t stream) {
  const int*   src = (const int*)d_in[0];
  const int*   dst = (const int*)d_in[1];
  const int*   gid = (const int*)d_in[2];
  const float* W0  = (const float*)d_in[3];
  const float* al0 = (const float*)d_in[4];
  const float* ar0 = (const float*)d_in[5];
  const float* W1  = (const float*)d_in[6];
  const float* al1 = (const float*)d_in[7];
  const float* ar1 = (const float*)d_in[8];
  const float* Wc  = (const float*)d_in[9];
  const float* bc  = (const float*)d_in[10];
  float* out = (float*)d_out;

  const int N = N_NODES, E = N_EDGES;
  float* ws  = (float*)d_ws;
  float* deg = ws;                         // [N]
  float* el  = deg + N;                    // [N*4]
  float* er  = el + N * 4;                 // [N*4]
  float* m   = er + N * 4;                 // [N*4]
  float* s   = m + N * 4;                  // [N*4]
  float* X   = s + N * 4;                  // [N*128] layer0 out -> h1 -> layer1 out -> h2
  float* z1  = X + (long long)N * NF;      // [N*128]
  float* hg  = z1 + (long long)N * NF;     // [B*128]
  float* cnt = hg + NB * NF;               // [B]

  const int B = 256;
  const long long ENF = (long long)E * NF;
  const int NNF = N * NF;

  // ---- in-degree -> layer0 node terms ----
  fill_f32<<<cdiv(N, B), B, 0, stream>>>(deg, 0.0f, N);
  k_degree<<<cdiv(E, B), B, 0, stream>>>(dst, deg, E);
  k_node0<<<cdiv(N * 4, B), B, 0, stream>>>(deg, W0, al0, ar0, el, er, N);

  // ---- layer0 softmax-aggregate ----
  fill_f32<<<cdiv(N * 4, B), B, 0, stream>>>(m, -3.0e38f, N * 4);
  fill_f32<<<cdiv(N * 4, B), B, 0, stream>>>(s, 0.0f, N * 4);
  fill_f32<<<cdiv(NNF, B), B, 0, stream>>>(X, 0.0f, NNF);
  k_edge_max<<<cdiv(E, B), B, 0, stream>>>(src, dst, el, er, m, E);
  k_edge_sum<<<cdiv(E, B), B, 0, stream>>>(src, dst, el, er, m, s, E);
  k_aggr0<<<cdiv(ENF, B), B, 0, stream>>>(src, dst, el, er, m, s, deg, W0, X, E);
  k_elu<<<cdiv(NNF, B), B, 0, stream>>>(X, NNF);      // X = h1

  // ---- layer1 projection (WMMA) + node terms ----
  {
    int ntiles = (N + 15) / 16;
    unsigned gblocks = ntiles < 1024 ? (unsigned)ntiles : 1024u;
    k_gemm_wmma<<<gblocks, 256, 0, stream>>>(X, W1, z1, N);
  }
  k_node1<<<cdiv(N * 4, B), B, 0, stream>>>(z1, al1, ar1, el, er, N);

  // ---- layer1 softmax-aggregate (reuse X as out1) ----
  fill_f32<<<cdiv(N * 4, B), B, 0, stream>>>(m, -3.0e38f, N * 4);
  fill_f32<<<cdiv(N * 4, B), B, 0, stream>>>(s, 0.0f, N * 4);
  fill_f32<<<cdiv(NNF, B), B, 0, stream>>>(X, 0.0f, NNF);
  k_edge_max<<<cdiv(E, B), B, 0, stream>>>(src, dst, el, er, m, E);
  k_edge_sum<<<cdiv(E, B), B, 0, stream>>>(src, dst, el, er, m, s, E);
  k_aggr1<<<cdiv(ENF, B), B, 0, stream>>>(src, dst, el, er, m, s, z1, X, E);
  k_elu<<<cdiv(NNF, B), B, 0, stream>>>(X, NNF);      // X = h2

  // ---- mean pool + classify ----
  fill_f32<<<cdiv(NB * NF, B), B, 0, stream>>>(hg, 0.0f, NB * NF);
  fill_f32<<<1, B, 0, stream>>>(cnt, 0.0f, NB);
  k_pool<<<cdiv(NNF, B), B, 0, stream>>>(X, gid, hg, cnt, N);
  k_classify<<<cdiv(NB * NCLS, 128), 128, 0, stream>>>(hg, cnt, Wc, bc, out);
}